// DetectionCircle_40106404610068
// MI455X (gfx1250) — compile-verified
//
#include <hip/hip_runtime.h>
#include <hip/hip_bf16.h>
#include <math.h>

typedef __attribute__((ext_vector_type(2))) float v2f;
typedef __attribute__((ext_vector_type(8))) float v8f;

#define GRID_G 104
#define CELLS 10816            // G*G
#define A_N 3
#define B_N 128
#define N_TOT (B_N * A_N * CELLS)   // 4,153,344
#define NBLOCKS 8192
#define EPSF 1e-9f
#define PIF 3.14159274101257f

__device__ __forceinline__ float sigm(float x) { return 1.0f / (1.0f + expf(-x)); }

// stable log(sigmoid(x)) = -log1p(exp(-x))
__device__ __forceinline__ float logsig(float x) {
    return (x >= 0.0f) ? -log1pf(expf(-x)) : (x - log1pf(expf(x)));
}

__device__ __forceinline__ float clamp11(float x) {
    return fminf(fmaxf(x, -1.0f), 1.0f);
}

__device__ __forceinline__ float riou_ratio(float trv, float arv) {
    float mn = fminf(trv, arv), mx = fmaxf(trv, arv);
    float q = mn / (mx + EPSF);
    return q * q;
}

__global__ __launch_bounds__(256)
void det_main_kernel(const float* __restrict__ x, const float* __restrict__ tg,
                     float* __restrict__ out,
                     float* __restrict__ wsD, float* __restrict__ wsC, float* __restrict__ wsF) {
    float sD = 0.0f, sC = 0.0f, sF = 0.0f;

    const float inv_g = 1.0f / 104.0f;
    for (int i = blockIdx.x * blockDim.x + threadIdx.x; i < N_TOT;
         i += gridDim.x * blockDim.x) {
        int cell = i % CELLS;            // gy*G + gx
        int t2   = i / CELLS;            // b*3 + a
        int gx   = cell % GRID_G;
        int gy   = cell / GRID_G;
        int a    = t2 % A_N;
        int b    = t2 / A_N;

        // x layout: [b][a*4 + c][gy][gx]; channel stride = CELLS
        int xb = t2 * (4 * CELLS) + cell;
        float p0 = x[xb];
        float p1 = x[xb + CELLS];
        float p2 = x[xb + 2 * CELLS];
        float p3 = x[xb + 3 * CELLS];

        // targets layout: [b][ch][gy][gx], ch = {tx, ty, tr, conf}
        int tb = b * (4 * CELLS) + cell;
        float tx  = tg[tb];
        float ty  = tg[tb + CELLS];
        float trr = tg[tb + 2 * CELLS];
        float cf  = tg[tb + 3 * CELLS];

        // anchor constants: pr = ANCHORS/stride (stride = 8), ar = ANCHORS/832
        float pra = (a == 0) ? 2.5f : ((a == 1) ? 6.25f : 13.75f);

        float bx = (sigm(p0) + (float)gx) * inv_g;
        float by = (sigm(p1) + (float)gy) * inv_g;
        float br = pra * expf(p2) * inv_g;
        float pc = sigm(p3);

        // output: [b][a*G*G + cell][4] == flat index i, 4 floats
        float4 o;
        o.x = bx * 832.0f;
        o.y = by * 832.0f;
        o.z = br * 832.0f;
        o.w = pc;
        reinterpret_cast<float4*>(out)[i] = o;

        // best anchor by radius IoU (argmax, first-max wins -> strict >)
        const float ar0 = 20.0f / 832.0f, ar1 = 50.0f / 832.0f, ar2 = 110.0f / 832.0f;
        int best = 0;
        float bv = riou_ratio(trr, ar0);
        float v1 = riou_ratio(trr, ar1);
        if (v1 > bv) { bv = v1; best = 1; }
        float v2 = riou_ratio(trr, ar2);
        if (v2 > bv) { best = 2; }

        bool obj = (best == a) && (cf > 0.5f);

        // circle DIoU between pred (bx,by,br) and target
        float tbx = ((float)gx + tx) * inv_g;
        float tby = ((float)gy + ty) * inv_g;
        float dx = bx - tbx, dy = by - tby;
        float d2 = dx * dx + dy * dy;
        float d  = sqrtf(d2 + EPSF);
        float r1 = br, r2 = trr;
        float rmin = fminf(r1, r2), rmax = fmaxf(r1, r2);
        float r1s = r1 * r1, r2s = r2 * r2;
        float a1 = clamp11((d2 + r1s - r2s) / (2.0f * d * r1 + EPSF));
        float a2 = clamp11((d2 + r2s - r1s) / (2.0f * d * r2 + EPSF));
        float tt = fmaxf((-d + r1 + r2) * (d + r1 - r2) * (d - r1 + r2) * (d + r1 + r2), 0.0f);
        float lens = r1s * acosf(a1) + r2s * acosf(a2) - 0.5f * sqrtf(tt);
        float inter = (d >= r1 + r2) ? 0.0f
                     : ((d <= rmax - rmin) ? PIF * rmin * rmin : lens);
        float uni = PIF * (r1s + r2s) - inter;
        float iou = inter / (uni + EPSF);
        float srr = d + r1 + r2;
        float pen = d2 / (srr * srr + EPSF);
        float diou = 1.0f - iou + pen;

        if (obj) { sD += diou; sC += 1.0f; }

        // focal loss (y is 0/1, so pick the active branch)
        float lp  = logsig(p3);
        float lnp = logsig(-p3);
        float one_m = 1.0f - pc;
        float fl = obj ? (-0.25f * one_m * one_m * lp)
                       : (-0.75f * pc * pc * lnp);
        sF += fl;
    }

    // wave32 reduction
    for (int m = 16; m > 0; m >>= 1) {
        sD += __shfl_xor(sD, m, 32);
        sC += __shfl_xor(sC, m, 32);
        sF += __shfl_xor(sF, m, 32);
    }
    __shared__ float red[3][8];
    int wave = threadIdx.x >> 5;
    int lane = threadIdx.x & 31;
    if (lane == 0) { red[0][wave] = sD; red[1][wave] = sC; red[2][wave] = sF; }
    __syncthreads();
    if (threadIdx.x == 0) {
        float aD = 0.0f, aC = 0.0f, aF = 0.0f;
        for (int w = 0; w < 8; ++w) { aD += red[0][w]; aC += red[1][w]; aF += red[2][w]; }
        wsD[blockIdx.x] = aD;
        wsC[blockIdx.x] = aC;
        wsF[blockIdx.x] = aF;
    }
}

// Final reduction of NBLOCKS partials per array using the f32 WMMA pipe:
// A = ones(16x4), B = 64-element data tile (4x16), C accumulates -> each of
// the 16 columns of D holds a running sum of 4 elements/iteration.
// One wave32, EXEC all ones (no divergence before the WMMAs).
__global__ __launch_bounds__(32)
void det_reduce_kernel(const float* __restrict__ wsD, const float* __restrict__ wsC,
                       const float* __restrict__ wsF, float* __restrict__ loss_out) {
    int l = threadIdx.x;
    int col  = l & 15;
    int half = l >> 4;

    v2f ones; ones.x = 1.0f; ones.y = 1.0f;

    const float* arrs[3] = { wsD, wsC, wsF };
    float sums[3];

    for (int k = 0; k < 3; ++k) {
        const float* arr = arrs[k];
        v8f acc = {};
        for (int base = 0; base < NBLOCKS; base += 64) {
            v2f bv;
            bv.x = arr[base + half * 32 + col];
            bv.y = arr[base + half * 32 + 16 + col];
            // D = ones(16x4) * B(4x16) + C  ->  column sums accumulate in C
            acc = __builtin_amdgcn_wmma_f32_16x16x4_f32(
                false, ones, false, bv, (short)0, acc, false, false);
        }
        // Row 0 (VGPR 0, lanes 0-15) and row 8 (lanes 16-31) both hold the
        // 16 column sums; fold 16 columns with shuffles.
        float s = acc[0];
        s += __shfl_xor(s, 1, 32);
        s += __shfl_xor(s, 2, 32);
        s += __shfl_xor(s, 4, 32);
        s += __shfl_xor(s, 8, 32);
        sums[k] = s;
    }

    if (l == 0) {
        float n_obj = fmaxf(sums[1], 1.0f);
        float loss = sums[0] / n_obj + sums[2] * (1.0f / (float)N_TOT);
        loss_out[0] = loss;
    }
}

extern "C" void kernel_launch(void* const* d_in, const int* in_sizes, int n_in,
                              void* d_out, int out_size, void* d_ws, size_t ws_size,
                              hipStream_t stream) {
    const float* x  = (const float*)d_in[0];
    const float* tg = (const float*)d_in[1];
    float* out = (float*)d_out;
    float* ws  = (float*)d_ws;

    float* wsD = ws;
    float* wsC = ws + NBLOCKS;
    float* wsF = ws + 2 * NBLOCKS;

    det_main_kernel<<<NBLOCKS, 256, 0, stream>>>(x, tg, out, wsD, wsC, wsF);

    // loss scalar lives right after the (B, A*G*G, 4) output block
    float* loss_out = out + (size_t)N_TOT * 4;
    det_reduce_kernel<<<1, 32, 0, stream>>>(wsD, wsC, wsF, loss_out);
}